// WaveletDenoise_6871947674026
// MI455X (gfx1250) — compile-verified
//
#include <hip/hip_runtime.h>
#include <hip/hip_bf16.h>
#include <math.h>

// ---------------------------------------------------------------------------
// sym8 wavelet denoise, 3 levels, for (B=16, T=8192, N=128) fp32.
// 2048 signals of length 8192. Memory-bound (~128MB io @ 23.3TB/s);
// conv tiles done as fp32 WMMA (V_WMMA_F32_16X16X4_F32) GEMMs.
// B (filter) fragments precomputed into LDS once per block so the inner
// loop is pure ds_load + v_wmma (no EXEC-masked constant fetches).
// ---------------------------------------------------------------------------

typedef __attribute__((ext_vector_type(2))) float v2f;
typedef __attribute__((ext_vector_type(8))) float v8f;

__constant__ float DEC_LO[16] = {
  -0.0033824159510061256f, -0.0005421323317911481f, 0.03169508781149298f,
   0.007607487324917605f,  -0.1432942383508097f,   -0.061273359067658524f,
   0.4813596512583722f,     0.7771857517005235f,    0.36444189483533895f,
  -0.05194583810770904f,   -0.027219029917056003f,  0.049137179673607506f,
   0.003808752013890615f,  -0.01495225833704823f,  -0.0003029205147241331f,
   0.0018899503327594609f };

// reference convolutions are correlations:
//   analysis uses h = [rec_lo, rec_hi], synthesis uses g = [dec_lo, dec_hi]
__device__ __forceinline__ float rec_lo_f(int j) {          // analysis lo
  return (j >= 0 && j < 16) ? DEC_LO[15 - j] : 0.f;
}
__device__ __forceinline__ float rec_hi_f(int j) {          // analysis hi: (-1)^j * dec_lo[j]
  return (j >= 0 && j < 16) ? (((j & 1) ? -1.f : 1.f) * DEC_LO[j]) : 0.f;
}
__device__ __forceinline__ float syn_lo_f(int j) {          // synthesis lo = dec_lo
  return (j >= 0 && j < 16) ? DEC_LO[j] : 0.f;
}
__device__ __forceinline__ float syn_hi_f(int j) {          // synthesis hi: (-1)^(j+1)*dec_lo[15-j]
  return (j >= 0 && j < 16) ? (((j & 1) ? 1.f : -1.f) * DEC_LO[15 - j]) : 0.f;
}

__device__ __forceinline__ v8f wmma4(v2f a, v2f b, v8f c) {
  // D = A(16x4,f32) * B(4x16,f32) + C(16x16,f32)
  return __builtin_amdgcn_wmma_f32_16x16x4_f32(false, a, false, b, (short)0, c,
                                               false, false);
}

// ---------------------------------------------------------------------------
// Analysis: lo[n] = sum_j x[2n+j-14] * h[j]   (symmetric reflect padding)
// Tile: 16 signals x 64 outputs per block (4 waves, 16x16 tile per wave).
// A(16x48) from LDS window, B(48x16) banded filter, 12 WMMA k-steps x2 filters.
// ---------------------------------------------------------------------------
__global__ __launch_bounds__(128) void wden_analysis(
    const float* __restrict__ in, float* __restrict__ lo_out,
    float* __restrict__ hi_out, int in_len, int in_stride,
    int out_len, int out_stride, int xin)
{
  __shared__ float tile[144][17];   // t-window x 16 signals (pad 17: no bank conflicts)
  __shared__ v2f   bloT[12 * 32];   // B fragments (lo filter), [kstep][lane]
  __shared__ v2f   bhiT[12 * 32];   // B fragments (hi filter)
  const int tid = threadIdx.x;
  const int N0  = blockIdx.x * 64;  // first output position of block
  const int s0  = blockIdx.y * 16;  // first signal of block

  // B-fragment tables: pure function of (kstep, lane); built once per block.
  for (int e = tid; e < 12 * 32; e += 128) {
    const int kk = e >> 5, ln = e & 31;
    const int kb = 4 * kk + 2 * (ln >> 4);        // K row for .x
    const int i0 = kb - 2 * (ln & 15);            // B[k][n] = h[k-2n]
    v2f bl; bl.x = rec_lo_f(i0); bl.y = rec_lo_f(i0 + 1);
    v2f bh; bh.x = rec_hi_f(i0); bh.y = rec_hi_f(i0 + 1);
    bloT[e] = bl;
    bhiT[e] = bh;
  }

  // cooperative, coalesced stage of input window t in [2*N0-14, 2*N0+129]
  for (int i = tid; i < 144 * 16; i += 128) {
    int r, c;
    if (xin) { r = i >> 4; c = i & 15; }          // x: channels contiguous
    else     { c = i / 144; r = i % 144; }        // ws: time contiguous
    int t = 2 * N0 - 14 + r;
    if (t < 0)        t = -t - 1;                 // symmetric reflect
    if (t >= in_len)  t = 2 * in_len - 1 - t;
    const int s = s0 + c;
    float v;
    if (xin) {
      v = in[((size_t)(s >> 7) * 8192 + (size_t)t) * 128 + (s & 127)];
    } else {
      v = in[(size_t)s * in_stride + t];
    }
    tile[r][c] = v;
  }
  __syncthreads();

  const int lane = tid & 31;
  const int w    = tid >> 5;        // wave id: output sub-tile
  const int half = lane >> 4;
  const int q    = lane & 15;       // A: row m (signal); B: col n (out pos)
  const int n0   = N0 + w * 16;
  const int rb   = 2 * (w * 16);    // LDS row base = 2*(n0-N0)

  v8f clo = {0.f,0.f,0.f,0.f,0.f,0.f,0.f,0.f};
  v8f chi = {0.f,0.f,0.f,0.f,0.f,0.f,0.f,0.f};
  #pragma unroll
  for (int kk = 0; kk < 12; ++kk) {
    const int kb = 4 * kk + 2 * half;             // global K row for a.x
    v2f a;  a.x = tile[rb + kb][q];  a.y = tile[rb + kb + 1][q];
    const v2f bl = bloT[kk * 32 + lane];          // ds_load_b64, conflict-free
    const v2f bh = bhiT[kk * 32 + lane];
    clo = wmma4(a, bl, clo);
    chi = wmma4(a, bh, chi);
  }

  const int pos = n0 + q;
  if (pos < out_len) {
    #pragma unroll
    for (int j = 0; j < 8; ++j) {                 // D row m = j + 8*half (signal)
      const size_t off = (size_t)(s0 + j + 8 * half) * out_stride + pos;
      lo_out[off] = clo[j];
      hi_out[off] = chi[j];
    }
  }
}

// ---------------------------------------------------------------------------
// Median -> threshold: exact rank-2051 of |cd1| per signal via 4-pass 8-bit
// radix select on the float bit pattern (monotone for non-negative floats).
// ---------------------------------------------------------------------------
__global__ __launch_bounds__(256) void wden_median(
    const float* __restrict__ cd1, float* __restrict__ thr,
    int n, int stride, int rank)
{
  __shared__ unsigned hist[256];
  __shared__ unsigned sh_prefix;
  __shared__ int      sh_rank;
  const int tid = threadIdx.x;
  const float* base = cd1 + (size_t)blockIdx.x * stride;

  unsigned prefix = 0u;
  int rk = rank;
  for (int p = 0; p < 4; ++p) {
    const int shift = 24 - 8 * p;
    const unsigned mask_hi = (p == 0) ? 0u : (0xFFFFFFFFu << (shift + 8));
    hist[tid] = 0u;
    __syncthreads();
    for (int i = tid; i < n; i += 256) {
      const unsigned key = __float_as_uint(fabsf(base[i]));
      if ((key & mask_hi) == prefix)
        atomicAdd(&hist[(key >> shift) & 255u], 1u);
    }
    __syncthreads();
    if (tid == 0) {
      unsigned cum = 0; int b = 0;
      for (; b < 255; ++b) {
        const unsigned h = hist[b];
        if (cum + h > (unsigned)rk) break;
        cum += h;
      }
      sh_rank   = rk - (int)cum;
      sh_prefix = prefix | ((unsigned)b << shift);
    }
    __syncthreads();
    prefix = sh_prefix;
    rk     = sh_rank;
    __syncthreads();
  }
  if (tid == 0) {
    const float med = __uint_as_float(prefix);
    const float lam = med / 0.6745f * sqrtf(2.0f * logf(8192.0f));
    thr[blockIdx.x] = 0.85f * lam;
  }
}

// ---------------------------------------------------------------------------
// Synthesis: y[t] = sum_a lo[t/2+a]*g0[2a+1]           (t even)
//                 + sum_a lo[(t-1)/2+a]*g0[2a]         (t odd)   (same for hi)
// Soft threshold fused into the hi LDS fill. 16 signals x 64 outputs / block.
// ---------------------------------------------------------------------------
__device__ __forceinline__ float syn_coef(int k, int dt, int hi) {
  int a, gi;
  if ((dt & 1) == 0) { a = k - (dt >> 1);       gi = 2 * a + 1; }
  else               { a = k - ((dt - 1) >> 1); gi = 2 * a;     }
  if (a < 0 || a > 7) return 0.f;
  return hi ? syn_hi_f(gi) : syn_lo_f(gi);
}

__global__ __launch_bounds__(128) void wden_synth(
    const float* __restrict__ loin, const float* __restrict__ hiin,
    const float* __restrict__ thr, int n, int lo_stride, int hi_stride,
    float* __restrict__ out, int out_len, int out_stride, int xout)
{
  __shared__ float ltile[40][17];
  __shared__ float htile[40][17];
  __shared__ float otile[64][17];
  __shared__ v2f   bslT[4 * 32];   // B fragments (lo path)
  __shared__ v2f   bshT[4 * 32];   // B fragments (hi path)
  const int tid = threadIdx.x;
  const int T0  = blockIdx.x * 64;   // first output t of block (even)
  const int s0  = blockIdx.y * 16;

  for (int e = tid; e < 4 * 32; e += 128) {
    const int kk = e >> 5, ln = e & 31;
    const int kb = 4 * kk + 2 * (ln >> 4);
    const int dt = ln & 15;
    v2f bl; bl.x = syn_coef(kb, dt, 0); bl.y = syn_coef(kb + 1, dt, 0);
    v2f bh; bh.x = syn_coef(kb, dt, 1); bh.y = syn_coef(kb + 1, dt, 1);
    bslT[e] = bl;
    bshT[e] = bh;
  }

  for (int i = tid; i < 40 * 16; i += 128) {
    const int c = i / 40, r = i % 40;
    const int idx = (T0 >> 1) + r;
    const int s = s0 + c;
    float lv = 0.f, hv = 0.f;
    if (idx < n) {
      lv = loin[(size_t)s * lo_stride + idx];
      const float h  = hiin[(size_t)s * hi_stride + idx];
      const float tv = thr[s];
      const float ah = fabsf(h) - tv;
      hv = (ah > 0.f) ? copysignf(ah, h) : 0.f;   // sign(h)*relu(|h|-thr)
    }
    ltile[r][c] = lv;
    htile[r][c] = hv;
  }
  __syncthreads();

  const int lane = tid & 31;
  const int w    = tid >> 5;
  const int half = lane >> 4;
  const int q    = lane & 15;

  v8f acc = {0.f,0.f,0.f,0.f,0.f,0.f,0.f,0.f};
  #pragma unroll
  for (int kk = 0; kk < 4; ++kk) {
    const int kb = 4 * kk + 2 * half;             // local K row
    v2f al; al.x = ltile[8 * w + kb][q]; al.y = ltile[8 * w + kb + 1][q];
    v2f ah; ah.x = htile[8 * w + kb][q]; ah.y = htile[8 * w + kb + 1][q];
    const v2f bl = bslT[kk * 32 + lane];
    const v2f bh = bshT[kk * 32 + lane];
    acc = wmma4(al, bl, acc);
    acc = wmma4(ah, bh, acc);
  }

  // D[m=signal][n=t] -> LDS transpose for coalesced stores
  #pragma unroll
  for (int j = 0; j < 8; ++j)
    otile[w * 16 + q][j + 8 * half] = acc[j];
  __syncthreads();

  if (xout) {   // final level: write x-layout (b,t,ch) contiguously along ch
    for (int i = tid; i < 64 * 16; i += 128) {
      const int tl = i >> 4, c = i & 15;
      const int s = s0 + c;
      out[((size_t)(s >> 7) * 8192 + (size_t)(T0 + tl)) * 128 + (s & 127)] =
          otile[tl][c];
    }
  } else {      // ws signal-major: contiguous along t
    for (int i = tid; i < 16 * 64; i += 128) {
      const int c = i >> 6, tl = i & 63;
      const int t = T0 + tl;
      if (t < out_len)
        out[(size_t)(s0 + c) * out_stride + t] = otile[tl][c];
    }
  }
}

// ---------------------------------------------------------------------------
extern "C" void kernel_launch(void* const* d_in, const int* in_sizes, int n_in,
                              void* d_out, int out_size, void* d_ws, size_t ws_size,
                              hipStream_t stream) {
  (void)in_sizes; (void)n_in; (void)out_size; (void)ws_size;
  const float* x = (const float*)d_in[0];
  float* out = (float*)d_out;
  float* ws  = (float*)d_ws;

  const int T = 8192, S = 2048;
  const int n1 = 4103, st1 = 4112;   // (8192+15)/2
  const int n2 = 2059, st2 = 2064;   // (4103+15)/2
  const int n3 = 1037, st3 = 1040;   // (2059+15)/2

  float* lo1 = ws;                          // reused as y1 during synthesis
  float* cd1 = lo1 + (size_t)S * st1;
  float* lo2 = cd1 + (size_t)S * st1;       // reused as y2 during synthesis
  float* cd2 = lo2 + (size_t)S * st2;
  float* lo3 = cd2 + (size_t)S * st2;
  float* cd3 = lo3 + (size_t)S * st3;
  float* thr = cd3 + (size_t)S * st3;       // total ~118 MB of ws

  const dim3 blk(128);
  // analysis cascade
  wden_analysis<<<dim3((n1 + 63) / 64, S / 16), blk, 0, stream>>>(
      x,   lo1, cd1, T,  0,   n1, st1, 1);
  wden_analysis<<<dim3((n2 + 63) / 64, S / 16), blk, 0, stream>>>(
      lo1, lo2, cd2, n1, st1, n2, st2, 0);
  wden_analysis<<<dim3((n3 + 63) / 64, S / 16), blk, 0, stream>>>(
      lo2, lo3, cd3, n2, st2, n3, st3, 0);
  // per-signal threshold from median(|cd1|), rank (n1-1)/2 = 2051
  wden_median<<<dim3(S), dim3(256), 0, stream>>>(cd1, thr, n1, st1, (n1 - 1) / 2);
  // synthesis cascade (y buffers alias consumed lo buffers)
  wden_synth<<<dim3((2 * n3 - 14 + 63) / 64, S / 16), blk, 0, stream>>>(
      lo3, cd3, thr, n3, st3, st3, lo2, 2 * n3 - 14, st2, 0);
  wden_synth<<<dim3((2 * n2 - 14 + 63) / 64, S / 16), blk, 0, stream>>>(
      lo2, cd2, thr, n2, st2, st2, lo1, 2 * n2 - 14, st1, 0);
  wden_synth<<<dim3(T / 64, S / 16), blk, 0, stream>>>(
      lo1, cd1, thr, n1, st1, st1, out, T, 0, 1);
}